// FixedPtPGD_70489003262097
// MI455X (gfx1250) — compile-verified
//
#include <hip/hip_runtime.h>
#include <hip/hip_bf16.h>

// FixedPtPGD on gfx1250: batched PGD with ADMM projection, restructured so the
// ADMM inner loop is ONE fp32 WMMA GEMM (1024x512x512) per iteration with the
// s/u/t updates fused into the GEMM epilogue. All matrices are L2-resident;
// fp32 WMMA (V_WMMA_F32_16X16X4_F32) matches reference precision exactly.
// GEMM uses a depth-4 software pipeline so loads stay ~3 k-steps ahead of the
// WMMAs that consume them (removes the s_wait_loadcnt 0x0 per-WMMA stalls).

typedef float v2f __attribute__((ext_vector_type(2)));
typedef float v8f __attribute__((ext_vector_type(8)));

namespace {
constexpr int BN = 1024;         // batch
constexpr int NN = 512;          // variable dim
constexpr int MI = 512;          // inequality constraints
constexpr int ME = 16;           // equality constraints
constexpr int KD = NN + ME;      // 528 KKT dim
constexpr int LDGJ = 2 * KD;     // 1056 augmented [K | I] row stride
constexpr float PALPHA = 0.05f;
constexpr int PGD_ITER = 3;
constexpr int ADMM_ITERS = 15;
}

__device__ __forceinline__ v8f wmma4(v2f a, v2f b, v8f c) {
  // D(16x16,f32) = A(16x4,f32) * B(4x16,f32) + C
  return __builtin_amdgcn_wmma_f32_16x16x4_f32(
      false, a, false, b, (short)0, c, false, false);
}

enum { MODE_PLAIN = 0, MODE_BIAS = 1, MODE_SINIT = 2, MODE_ITER = 3, MODE_FINALZ = 4 };

// C[M x N] = A[M x K] . B[N x K]^T  (row-major operands, dot along K).
// 16(M) x 64(N) tile per wave; K compile-time (always 512 here); depth-4
// software-pipelined operand fetch.
template <int MODE, int K>
__global__ __launch_bounds__(256) void wmma_gemm_abt(
    const float* __restrict__ A, int lda,
    const float* __restrict__ B, int ldb,
    int tilesN,
    float* __restrict__ out0, float* __restrict__ out1,
    const float* __restrict__ e0,   // col-vector: zc / h   (len N)
    const float* __restrict__ e1,   // matrix: xg / xz      (M x ldc)
    const float* __restrict__ e2,   // matrix: u            (M x ldc)
    int ldc) {
  const int wave = blockIdx.x * 8 + (threadIdx.x >> 5);
  const int lane = threadIdx.x & 31;
  const int tm = wave / tilesN, tn = wave % tilesN;
  const int m0 = tm * 16, n0 = tn * 64;
  const int r = lane & 15, half = lane >> 4;

  // Operand addressing per documented 32-bit 16x4 layout:
  // lane r (0..15) holds row m0+r / n0+..; half selects K pair {0,1} vs {2,3}.
  const float* ap  = A + (size_t)(m0 + r) * lda + 2 * half;
  const float* bp0 = B + (size_t)(n0 + 0 + r) * ldb + 2 * half;
  const float* bp1 = B + (size_t)(n0 + 16 + r) * ldb + 2 * half;
  const float* bp2 = B + (size_t)(n0 + 32 + r) * ldb + 2 * half;
  const float* bp3 = B + (size_t)(n0 + 48 + r) * ldb + 2 * half;

  const v8f vzero = {0.f, 0.f, 0.f, 0.f, 0.f, 0.f, 0.f, 0.f};
  v8f acc[4] = {vzero, vzero, vzero, vzero};

  // Depth-4 pipeline: slots hold k-steps kb, kb+4, kb+8, kb+12.
  v2f As[4], Bs[4][4];
#pragma unroll
  for (int s = 0; s < 4; ++s) {
    const int k = 4 * s;
    As[s]    = *(const v2f*)(ap + k);
    Bs[s][0] = *(const v2f*)(bp0 + k);
    Bs[s][1] = *(const v2f*)(bp1 + k);
    Bs[s][2] = *(const v2f*)(bp2 + k);
    Bs[s][3] = *(const v2f*)(bp3 + k);
  }
  for (int kb = 0; kb <= K - 32; kb += 16) {
#pragma unroll
    for (int s = 0; s < 4; ++s) {
      acc[0] = wmma4(As[s], Bs[s][0], acc[0]);
      acc[1] = wmma4(As[s], Bs[s][1], acc[1]);
      acc[2] = wmma4(As[s], Bs[s][2], acc[2]);
      acc[3] = wmma4(As[s], Bs[s][3], acc[3]);
      const int kn = kb + 16 + 4 * s;   // always < K in this loop
      As[s]    = *(const v2f*)(ap + kn);
      Bs[s][0] = *(const v2f*)(bp0 + kn);
      Bs[s][1] = *(const v2f*)(bp1 + kn);
      Bs[s][2] = *(const v2f*)(bp2 + kn);
      Bs[s][3] = *(const v2f*)(bp3 + kn);
    }
  }
#pragma unroll
  for (int s = 0; s < 4; ++s) {        // drain last block (kb = K-16)
    acc[0] = wmma4(As[s], Bs[s][0], acc[0]);
    acc[1] = wmma4(As[s], Bs[s][1], acc[1]);
    acc[2] = wmma4(As[s], Bs[s][2], acc[2]);
    acc[3] = wmma4(As[s], Bs[s][3], acc[3]);
  }

  // Epilogue. D layout: VGPR r2 holds row m0+8*half+r2, col n0+16*nj+r.
#pragma unroll
  for (int nj = 0; nj < 4; ++nj) {
#pragma unroll
    for (int r2 = 0; r2 < 8; ++r2) {
      const int m = m0 + 8 * half + r2;
      const int n = n0 + 16 * nj + r;
      const size_t idx = (size_t)m * ldc + n;
      const float v = acc[nj][r2];
      if (MODE == MODE_PLAIN) {
        out0[idx] = v;
      } else if (MODE == MODE_BIAS) {
        out0[idx] = v + e0[n];               // xz = x.KinvUL^T + zc
      } else if (MODE == MODE_SINIT) {
        // Gx = v; s0 = max(0, h-Gx); u0 = 0; store t0 = h-s0-u0 = min(h,Gx)
        out0[idx] = fminf(e0[n], v);
        out1[idx] = 0.f;
      } else if (MODE == MODE_ITER) {
        const float Gz = e1[idx] + v;        // Gz = xg + t.R
        const float hh = e0[n];
        const float uu = e2[idx];
        const float sn = fmaxf(0.f, hh - Gz - uu);
        const float un = uu + Gz + sn - hh;
        out0[idx] = hh - sn - un;            // t_next
        out1[idx] = un;                      // u (in-place: 1 lane per elem)
      } else {                               // MODE_FINALZ
        out0[idx] = e1[idx] + v;             // z = xz + t.Q
      }
    }
  }
}

// Build augmented [K | I], K = [[I + G^T G, A^T],[A, 0]]  (RHO == 1).
__global__ __launch_bounds__(256) void buildK_kernel(
    const float* __restrict__ G, const float* __restrict__ Aeq,
    float* __restrict__ gj) {
  const int idx = blockIdx.x * 256 + threadIdx.x;
  if (idx >= KD * KD) return;
  const int i = idx / KD, j = idx % KD;
  float val;
  if (i < NN && j < NN) {
    float acc = (i == j) ? 1.f : 0.f;
    for (int m = 0; m < MI; ++m) acc += G[(size_t)m * NN + i] * G[(size_t)m * NN + j];
    val = acc;
  } else if (i < NN) {
    val = Aeq[(size_t)(j - NN) * NN + i];
  } else if (j < NN) {
    val = Aeq[(size_t)(i - NN) * NN + j];
  } else {
    val = 0.f;
  }
  gj[(size_t)i * LDGJ + j] = val;
  gj[(size_t)i * LDGJ + KD + j] = (i == j) ? 1.f : 0.f;
}

// One Gauss-Jordan pivot step, ping-pong src->dst (no pivoting needed: leading
// block is SPD, trailing Schur complement negative definite).
__global__ __launch_bounds__(256) void gj_pivot_kernel(
    const float* __restrict__ src, float* __restrict__ dst, int p) {
  const int idx = blockIdx.x * 256 + threadIdx.x;
  if (idx >= KD * LDGJ) return;
  const int i = idx / LDGJ, j = idx % LDGJ;
  const float piv = src[(size_t)p * LDGJ + p];
  const float prow = src[(size_t)p * LDGJ + j] / piv;
  dst[idx] = (i == p) ? prow : src[idx] - src[(size_t)i * LDGJ + p] * prow;
}

// zc[i] = sum_e b[e] * Kinv[i, NN+e]
__global__ __launch_bounds__(256) void zc_kernel(
    const float* __restrict__ gj, const float* __restrict__ bvec,
    float* __restrict__ zc) {
  const int i = blockIdx.x * 256 + threadIdx.x;
  if (i >= NN) return;
  float acc = 0.f;
  for (int e = 0; e < ME; ++e)
    acc += bvec[e] * gj[(size_t)i * LDGJ + KD + NN + e];
  zc[i] = acc;
}

// x = (1-alpha)*x + alpha*c  (PGD gradient step, in place)
__global__ __launch_bounds__(256) void pgd_step_kernel(
    float* __restrict__ x, const float* __restrict__ c, int n) {
  const int i = blockIdx.x * 256 + threadIdx.x;
  if (i < n) x[i] = (1.f - PALPHA) * x[i] + PALPHA * c[i];
}

// ---------------- host side ----------------

template <int MODE>
static void launch_gemm(hipStream_t st, const float* A, int lda, const float* B,
                        int ldb, int M, int N, float* o0, float* o1,
                        const float* e0, const float* e1, const float* e2,
                        int ldc) {
  const int waves = (M / 16) * (N / 64);
  wmma_gemm_abt<MODE, 512><<<waves / 8, 256, 0, st>>>(A, lda, B, ldb, N / 64,
                                                      o0, o1, e0, e1, e2, ldc);
}

extern "C" void kernel_launch(void* const* d_in, const int* in_sizes, int n_in,
                              void* d_out, int out_size, void* d_ws,
                              size_t ws_size, hipStream_t stream) {
  const float* c = (const float*)d_in[0];     // [BN, NN]
  const float* G = (const float*)d_in[1];     // [MI, NN]
  const float* h = (const float*)d_in[2];     // [MI]
  const float* Aeq = (const float*)d_in[3];   // [ME, NN]
  const float* bvec = (const float*)d_in[4];  // [ME]
  float* out = (float*)d_out;                 // [BN, NN]

  float* w = (float*)d_ws;
  size_t o = 0;
  float* gjA = w + o; o += (size_t)KD * LDGJ;
  float* gjB = w + o; o += (size_t)KD * LDGJ;
  float* Q   = w + o; o += (size_t)MI * NN;   // Q[m,i]  = sum_j G[m,j] Kinv[i,j]
  float* Qt  = w + o; o += (size_t)NN * MI;   // Qt[i,m] = Q[m,i]
  float* Rt  = w + o; o += (size_t)MI * MI;   // Rt[m',m] = sum_i Q[m,i] G[m',i]
  float* zc  = w + o; o += NN;
  float* xz  = w + o; o += (size_t)BN * NN;
  float* xg  = w + o; o += (size_t)BN * MI;
  float* tA  = w + o; o += (size_t)BN * MI;
  float* tB  = w + o; o += (size_t)BN * MI;
  float* u   = w + o; o += (size_t)BN * MI;
  float* xb  = w + o; o += (size_t)BN * NN;

  const float* KinvUL = gjA + KD;  // Kinv[i,j], row stride LDGJ (right half)

  // ---- one-time precompute ----
  buildK_kernel<<<(KD * KD + 255) / 256, 256, 0, stream>>>(G, Aeq, gjA);
  {
    float* src = gjA;
    float* dst = gjB;
    for (int p = 0; p < KD; ++p) {  // 528 even -> result lands back in gjA
      gj_pivot_kernel<<<(KD * LDGJ + 255) / 256, 256, 0, stream>>>(src, dst, p);
      float* tmp = src; src = dst; dst = tmp;
    }
  }
  zc_kernel<<<(NN + 255) / 256, 256, 0, stream>>>(gjA, bvec, zc);
  // Q = G . KinvUL^T ; Qt = KinvUL . G^T ; Rt = G . Q^T
  launch_gemm<MODE_PLAIN>(stream, G, NN, KinvUL, LDGJ, MI, NN, Q, nullptr,
                          nullptr, nullptr, nullptr, NN);
  launch_gemm<MODE_PLAIN>(stream, KinvUL, LDGJ, G, NN, NN, MI, Qt, nullptr,
                          nullptr, nullptr, nullptr, MI);
  launch_gemm<MODE_PLAIN>(stream, G, NN, Q, NN, MI, MI, Rt, nullptr,
                          nullptr, nullptr, nullptr, MI);

  // ---- projection: zdst = argmin ||z - xin||^2 s.t. Gz<=h, Az=b ----
  auto project = [&](const float* xin, float* zdst) {
    // xz = xin.KinvUL^T + zc ; t0 = min(h, xin.G^T), u0 = 0 ; xg = xz.G^T
    launch_gemm<MODE_BIAS>(stream, xin, NN, KinvUL, LDGJ, BN, NN, xz,
                           nullptr, zc, nullptr, nullptr, NN);
    launch_gemm<MODE_SINIT>(stream, xin, NN, G, NN, BN, MI, tA, u, h,
                            nullptr, nullptr, MI);
    launch_gemm<MODE_PLAIN>(stream, xz, NN, G, NN, BN, MI, xg, nullptr,
                            nullptr, nullptr, nullptr, MI);
    float* tcur = tA;
    float* tnxt = tB;
    for (int it = 0; it < ADMM_ITERS; ++it) {
      // Gz = xg + t.Rt^T ; s' = max(0,h-Gz-u); u' = u+Gz+s'-h; t' = h-s'-u'
      launch_gemm<MODE_ITER>(stream, tcur, MI, Rt, MI, BN, MI, tnxt, u, h,
                             xg, u, MI);
      float* tmp = tcur; tcur = tnxt; tnxt = tmp;
    }
    // z = xz + t.Qt^T
    launch_gemm<MODE_FINALZ>(stream, tcur, MI, Qt, MI, BN, NN, zdst,
                             nullptr, nullptr, xz, nullptr, NN);
  };

  project(c, xb);  // x = project(c)
  for (int i = 0; i < PGD_ITER; ++i) {
    pgd_step_kernel<<<(BN * NN + 255) / 256, 256, 0, stream>>>(xb, c, BN * NN);
    project(xb, (i == PGD_ITER - 1) ? out : xb);
  }
}